// AggGCNConv_30227979829558
// MI455X (gfx1250) — compile-verified
//
#include <hip/hip_runtime.h>
#include <hip/hip_bf16.h>

typedef __attribute__((ext_vector_type(2))) float v2f;
typedef __attribute__((ext_vector_type(8))) float v8f;

#define F_IN 128
#define HID  64
#define CLS  16

// ---------------------------------------------------------------------------
// Degree / normalization kernels
// ---------------------------------------------------------------------------
__global__ void deg_init_kernel(float* __restrict__ deg, int n) {
    int i = blockIdx.x * 256 + threadIdx.x;
    if (i < n) deg[i] = 1.0f;              // self-loop counts in degree
}

__global__ void deg_accum_kernel(const int* __restrict__ dst,
                                 float* __restrict__ deg, int e) {
    int i = blockIdx.x * 256 + threadIdx.x;
    if (i < e) atomicAdd(&deg[dst[i]], 1.0f);
}

__global__ void dinv_kernel(const float* __restrict__ deg,
                            float* __restrict__ dinv, int n) {
    int i = blockIdx.x * 256 + threadIdx.x;
    if (i < n) dinv[i] = rsqrtf(deg[i]);
}

// ---------------------------------------------------------------------------
// GEMM1: h1[N x 64] = x[N x 128] @ W1[128 x 64]   (fp32 WMMA 16x16x4)
// One wave per 16-row tile; computes the full 16x64 strip (4 n-tiles).
// W1 staged in LDS in paired-K layout: lds[p][n] = {W[2p][n], W[2p+1][n]}
// so each B fragment is ONE aligned ds_load_b64 (no repack movs).
// Requires N % 16 == 0 (true: 100000 = 6250*16).
// ---------------------------------------------------------------------------
__global__ __launch_bounds__(256) void gemm1_wmma_kernel(
    const float* __restrict__ x, const float* __restrict__ W1,
    float* __restrict__ h1, int nTiles)
{
    __shared__ v2f w[(F_IN / 2) * HID];    // 64*64 v2f = 32 KB
    for (int i = threadIdx.x; i < F_IN * HID; i += 256) {
        int k = i / HID, n = i % HID;
        w[(k >> 1) * HID + n][k & 1] = W1[i];
    }
    __syncthreads();

    const int wave = threadIdx.x >> 5;
    const int lane = threadIdx.x & 31;
    const int tile = blockIdx.x * 8 + wave;
    if (tile >= nTiles) return;            // wave-uniform: EXEC all-ones inside

    const int lhalf = lane >> 4;           // 0: K pair {0,1}; 1: K pair {2,3}
    const int l16   = lane & 15;
    const float* xrow = x + (long)(tile * 16 + l16) * F_IN;   // A: row = M

    v8f acc0 = {}, acc1 = {}, acc2 = {}, acc3 = {};
    for (int k = 0; k < F_IN; k += 4) {
        const int ka = k + 2 * lhalf;      // even
        v2f a = *(const v2f*)(xrow + ka);  // 8B-aligned global_load_b64
        const v2f* wr = &w[(ka >> 1) * HID + l16];
        v2f b0 = wr[0];
        v2f b1 = wr[16];
        v2f b2 = wr[32];
        v2f b3 = wr[48];
        acc0 = __builtin_amdgcn_wmma_f32_16x16x4_f32(false, a, false, b0, (short)0, acc0, false, false);
        acc1 = __builtin_amdgcn_wmma_f32_16x16x4_f32(false, a, false, b1, (short)0, acc1, false, false);
        acc2 = __builtin_amdgcn_wmma_f32_16x16x4_f32(false, a, false, b2, (short)0, acc2, false, false);
        acc3 = __builtin_amdgcn_wmma_f32_16x16x4_f32(false, a, false, b3, (short)0, acc3, false, false);
    }

    // C/D layout: VGPR r -> row = r + 8*lhalf, col = l16 (+16*ntile)
    float* out = h1 + ((long)tile * 16 + 8 * lhalf) * HID + l16;
#pragma unroll
    for (int r = 0; r < 8; ++r) {
        out[(long)r * HID + 0]  = acc0[r];
        out[(long)r * HID + 16] = acc1[r];
        out[(long)r * HID + 32] = acc2[r];
        out[(long)r * HID + 48] = acc3[r];
    }
}

// ---------------------------------------------------------------------------
// GEMM2: h2[N x 16] = relu(agg1)[N x 64] @ W2[64 x 16]   (fp32 WMMA)
// ReLU fused into the A-fragment load. Same paired-K LDS layout for W2.
// ---------------------------------------------------------------------------
__global__ __launch_bounds__(256) void gemm2_wmma_kernel(
    const float* __restrict__ agg1, const float* __restrict__ W2,
    float* __restrict__ h2, int nTiles)
{
    __shared__ v2f w[(HID / 2) * CLS];     // 32*16 v2f = 4 KB
    for (int i = threadIdx.x; i < HID * CLS; i += 256) {
        int k = i / CLS, n = i % CLS;
        w[(k >> 1) * CLS + n][k & 1] = W2[i];
    }
    __syncthreads();

    const int wave = threadIdx.x >> 5;
    const int lane = threadIdx.x & 31;
    const int tile = blockIdx.x * 8 + wave;
    if (tile >= nTiles) return;

    const int lhalf = lane >> 4;
    const int l16   = lane & 15;
    const float* arow = agg1 + (long)(tile * 16 + l16) * HID;

    v8f acc = {};
    for (int k = 0; k < HID; k += 4) {
        const int ka = k + 2 * lhalf;
        v2f a = *(const v2f*)(arow + ka);
        a.x = fmaxf(a.x, 0.0f);
        a.y = fmaxf(a.y, 0.0f);
        v2f b = w[(ka >> 1) * CLS + l16];
        acc = __builtin_amdgcn_wmma_f32_16x16x4_f32(false, a, false, b, (short)0, acc, false, false);
    }

    float* out = h2 + ((long)tile * 16 + 8 * lhalf) * CLS + l16;
#pragma unroll
    for (int r = 0; r < 8; ++r) out[r * CLS] = acc[r];
}

// ---------------------------------------------------------------------------
// Self-loop + bias initialization: agg[i][f] = h[i][f] * (1/deg_i) + b[f]
// (1/deg_i == dinv_i^2). One thread per (node, feature).
// ---------------------------------------------------------------------------
template <int FDIM>
__global__ void agg_init_kernel(const float* __restrict__ h,
                                const float* __restrict__ dinv,
                                const float* __restrict__ bias,
                                float* __restrict__ agg, int total)
{
    int i = blockIdx.x * 256 + threadIdx.x;
    if (i >= total) return;
    int node = i / FDIM;
    int f    = i % FDIM;
    float di = dinv[node];
    agg[i] = h[i] * (di * di) + bias[f];
}

// ---------------------------------------------------------------------------
// Edge scatter: agg[dst][f] += h[src][f] * dinv[src] * dinv[dst]
// Consecutive threads -> same edge, consecutive features: coalesced gather
// and coalesced atomic adds. All data L2-resident (h/agg < 52 MB << 192 MB).
// ---------------------------------------------------------------------------
template <int FDIM>
__global__ void edge_agg_kernel(const int* __restrict__ src,
                                const int* __restrict__ dst,
                                const float* __restrict__ h,
                                const float* __restrict__ dinv,
                                float* __restrict__ agg, int e)
{
    int gid = blockIdx.x * 256 + threadIdx.x;
    int eidx = gid / FDIM;
    int f    = gid % FDIM;
    if (eidx >= e) return;
    int s = src[eidx];
    int d = dst[eidx];
    float norm = dinv[s] * dinv[d];
    atomicAdd(&agg[(long)d * FDIM + f], h[(long)s * FDIM + f] * norm);
}

// ---------------------------------------------------------------------------
// Final: out = log_softmax(relu(agg2)) over 16 classes. One thread per node.
// ---------------------------------------------------------------------------
__global__ void logsoftmax_kernel(const float* __restrict__ agg2,
                                  float* __restrict__ out, int n)
{
    int i = blockIdx.x * 256 + threadIdx.x;
    if (i >= n) return;
    const float* z = agg2 + (long)i * CLS;
    float v[CLS];
    float m = -3.402823e38f;
#pragma unroll
    for (int c = 0; c < CLS; ++c) { v[c] = fmaxf(z[c], 0.0f); m = fmaxf(m, v[c]); }
    float s = 0.0f;
#pragma unroll
    for (int c = 0; c < CLS; ++c) s += expf(v[c] - m);
    float l = m + logf(s);
    float* o = out + (long)i * CLS;
#pragma unroll
    for (int c = 0; c < CLS; ++c) o[c] = v[c] - l;
}

// ---------------------------------------------------------------------------
// Launcher
// ---------------------------------------------------------------------------
extern "C" void kernel_launch(void* const* d_in, const int* in_sizes, int n_in,
                              void* d_out, int out_size, void* d_ws, size_t ws_size,
                              hipStream_t stream)
{
    const float* x  = (const float*)d_in[0];
    const int*   ei = (const int*)  d_in[1];
    const float* W1 = (const float*)d_in[2];
    const float* b1 = (const float*)d_in[3];
    const float* W2 = (const float*)d_in[4];
    const float* b2 = (const float*)d_in[5];

    const int N = in_sizes[0] / F_IN;     // 100000 (divisible by 16)
    const int E = in_sizes[1] / 2;        // 1600000
    const int* src = ei;
    const int* dst = ei + E;

    // Workspace layout (floats): h1 | agg1 | h2 | agg2 | deg | dinv  (~65 MB)
    float* ws   = (float*)d_ws;
    float* h1   = ws;
    float* agg1 = h1   + (size_t)N * HID;
    float* h2   = agg1 + (size_t)N * HID;
    float* agg2 = h2   + (size_t)N * CLS;
    float* deg  = agg2 + (size_t)N * CLS;
    float* dinv = deg  + (size_t)N;

    const int nTiles = N / 16;                         // 6250
    const int gemmBlocks = (nTiles + 7) / 8;           // 8 waves/block

    // 1) degrees (+self loop) and d^-1/2
    deg_init_kernel<<<(N + 255) / 256, 256, 0, stream>>>(deg, N);
    deg_accum_kernel<<<(E + 255) / 256, 256, 0, stream>>>(dst, deg, E);
    dinv_kernel<<<(N + 255) / 256, 256, 0, stream>>>(deg, dinv, N);

    // 2) layer 1: GEMM -> self-term+bias -> edge scatter
    gemm1_wmma_kernel<<<gemmBlocks, 256, 0, stream>>>(x, W1, h1, nTiles);
    {
        int total = N * HID;
        agg_init_kernel<HID><<<(total + 255) / 256, 256, 0, stream>>>(h1, dinv, b1, agg1, total);
        long work = (long)E * HID;
        edge_agg_kernel<HID><<<(int)((work + 255) / 256), 256, 0, stream>>>(src, dst, h1, dinv, agg1, E);
    }

    // 3) layer 2: GEMM (relu fused) -> self-term+bias -> edge scatter
    gemm2_wmma_kernel<<<gemmBlocks, 256, 0, stream>>>(agg1, W2, h2, nTiles);
    {
        int total = N * CLS;
        agg_init_kernel<CLS><<<(total + 255) / 256, 256, 0, stream>>>(h2, dinv, b2, agg2, total);
        long work = (long)E * CLS;
        edge_agg_kernel<CLS><<<(int)((work + 255) / 256), 256, 0, stream>>>(src, dst, h2, dinv, agg2, E);
    }

    // 4) relu + log_softmax -> d_out
    logsoftmax_kernel<<<(N + 255) / 256, 256, 0, stream>>>(agg2, (float*)d_out, N);
}